// Block_584115552374
// MI455X (gfx1250) — compile-verified
//
#include <hip/hip_runtime.h>
#include <hip/hip_bf16.h>
#include <cstdint>

// Graph-transformer layer for MI455X (gfx1250, wave32).
// N=10000 nodes, E=320000 edges, D=256, H=4 heads, HD=64.

#define NN 10000
#define EE 320000
#define DD 256
#define HH 4
#define HD 64

typedef __attribute__((ext_vector_type(2))) float v2f;
typedef __attribute__((ext_vector_type(8))) float v8f;

#define ASTR 260   // A panel LDS stride (256 + 4): rows on distinct bank groups
#define WSTR 272   // W panel LDS stride (16*17): K-rows 16 banks apart

// ---------------------------------------------------------------------------
// gfx1250 async global->LDS copy (ASYNCcnt-tracked, bypasses VGPRs).
// ---------------------------------------------------------------------------
__device__ __forceinline__ void async_ld_b128(unsigned lds_off, const float* g) {
  asm volatile("global_load_async_to_lds_b128 %0, %1, off"
               :: "v"(lds_off), "v"(g)
               : "memory");
}
__device__ __forceinline__ void wait_async0() {
  asm volatile("s_wait_asynccnt 0x0" ::: "memory");
}
__device__ __forceinline__ unsigned lds_addr_of(const void* p) {
  // addrspace(3) offsets live in the low 32 bits of the generic pointer
  return (unsigned)(uintptr_t)p;
}

// ---------------------------------------------------------------------------
// LayerNorm (pre-norm): one block (256 threads) per row.
// ---------------------------------------------------------------------------
__global__ __launch_bounds__(256)
void ln1_kernel(const float* __restrict__ x, const float* __restrict__ g,
                const float* __restrict__ b, float* __restrict__ y) {
  __shared__ float red[256];
  const int row = blockIdx.x;
  const int tid = threadIdx.x;
  const float v = x[row * DD + tid];

  red[tid] = v;
  __syncthreads();
  #pragma unroll
  for (int s = 128; s > 0; s >>= 1) {
    if (tid < s) red[tid] += red[tid + s];
    __syncthreads();
  }
  const float mean = red[0] * (1.0f / 256.0f);
  __syncthreads();

  const float d = v - mean;
  red[tid] = d * d;
  __syncthreads();
  #pragma unroll
  for (int s = 128; s > 0; s >>= 1) {
    if (tid < s) red[tid] += red[tid + s];
    __syncthreads();
  }
  const float var = red[0] * (1.0f / 256.0f);
  const float inv = rsqrtf(var + 1e-5f);
  y[row * DD + tid] = d * inv * g[tid] + b[tid];
}

// ---------------------------------------------------------------------------
// Fused: out_final = residual + relu(layer_norm(t, g2, b2))
// ---------------------------------------------------------------------------
__global__ __launch_bounds__(256)
void ln2_kernel(const float* __restrict__ t, const float* __restrict__ resid,
                const float* __restrict__ g, const float* __restrict__ b,
                float* __restrict__ y) {
  __shared__ float red[256];
  const int row = blockIdx.x;
  const int tid = threadIdx.x;
  const float v = t[row * DD + tid];

  red[tid] = v;
  __syncthreads();
  #pragma unroll
  for (int s = 128; s > 0; s >>= 1) {
    if (tid < s) red[tid] += red[tid + s];
    __syncthreads();
  }
  const float mean = red[0] * (1.0f / 256.0f);
  __syncthreads();

  const float d = v - mean;
  red[tid] = d * d;
  __syncthreads();
  #pragma unroll
  for (int s = 128; s > 0; s >>= 1) {
    if (tid < s) red[tid] += red[tid + s];
    __syncthreads();
  }
  const float var = red[0] * (1.0f / 256.0f);
  const float inv = rsqrtf(var + 1e-5f);
  const float ln  = d * inv * g[tid] + b[tid];
  y[row * DD + tid] = resid[row * DD + tid] + fmaxf(ln, 0.0f);
}

// ---------------------------------------------------------------------------
// GEMM: Y[16r x 256] = A[16r x 256] @ W[256 x 256] + bias, via
// V_WMMA_F32_16X16X4_F32 (fp32 exact). 8 waves/block.
// A panel and double-buffered 16-row W chunks are staged in LDS with
// GLOBAL_LOAD_ASYNC_TO_LDS_B128; compute reads only LDS.
// A frag (16x4):  lanes 0-15 -> K = k0..k0+1, lanes 16-31 -> K = k0+2..k0+3
// B frag (4x16):  lane n (mod 16) holds column n for the matching K pair
// C/D (16x16):    VGPR i -> row i (lanes 0-15) / row i+8 (lanes 16-31)
// ---------------------------------------------------------------------------
__global__ __launch_bounds__(256)
void gemm16_kernel(const float* __restrict__ A, const float* __restrict__ W,
                   const float* __restrict__ bias, float* __restrict__ Y) {
  __shared__ float As[16 * ASTR];
  __shared__ float Wsb[2][16 * WSTR];
  const int tid  = threadIdx.x;
  const int row0 = blockIdx.x * 16;

  const unsigned asBase  = lds_addr_of(&As[0]);
  const unsigned wsBase0 = lds_addr_of(&Wsb[0][0]);
  const unsigned wsBase1 = lds_addr_of(&Wsb[1][0]);

  // A panel: 16 rows x 256 cols = 1024 b128 slots, 4 per thread.
  #pragma unroll
  for (int i = 0; i < 4; ++i) {
    const int slot = i * 256 + tid;
    const int r  = slot >> 6;          // row 0..15
    const int cg = slot & 63;          // 4-float column group
    async_ld_b128(asBase + (unsigned)(r * ASTR + cg * 4) * 4u,
                  A + (row0 + r) * DD + cg * 4);
  }
  // W chunk 0 (K rows 0..15, all 256 columns)
  #pragma unroll
  for (int i = 0; i < 4; ++i) {
    const int slot = i * 256 + tid;
    const int r  = slot >> 6;
    const int cg = slot & 63;
    async_ld_b128(wsBase0 + (unsigned)(r * WSTR + cg * 4) * 4u,
                  W + r * DD + cg * 4);
  }
  wait_async0();
  __syncthreads();

  const int wave = tid >> 5;
  const int lane = tid & 31;
  const int r16  = lane & 15;
  const int kk   = (lane >> 4) * 2;    // 0 for lanes 0-15, 2 for lanes 16-31
  const int n0a  = (wave * 2    ) * 16;
  const int n0b  = (wave * 2 + 1) * 16;

  v8f acc0 = {}, acc1 = {};

  for (int kc = 0; kc < 16; ++kc) {
    // Prefetch next W chunk into the other buffer (overlaps with WMMAs).
    if (kc < 15) {
      const unsigned nxtW = (kc & 1) ? wsBase0 : wsBase1;
      const float* Wg = W + (kc + 1) * 16 * DD;
      #pragma unroll
      for (int i = 0; i < 4; ++i) {
        const int slot = i * 256 + tid;
        const int r  = slot >> 6;
        const int cg = slot & 63;
        async_ld_b128(nxtW + (unsigned)(r * WSTR + cg * 4) * 4u,
                      Wg + r * DD + cg * 4);
      }
    }

    const float* __restrict__ Wc = &Wsb[kc & 1][0];
    #pragma unroll
    for (int kl = 0; kl < 16; kl += 4) {
      v2f a = *(const v2f*)&As[r16 * ASTR + kc * 16 + kl + kk];
      v2f bA, bB;
      bA[0] = Wc[(kl + kk    ) * WSTR + n0a + r16];
      bA[1] = Wc[(kl + kk + 1) * WSTR + n0a + r16];
      acc0 = __builtin_amdgcn_wmma_f32_16x16x4_f32(
          false, a, false, bA, (short)0, acc0, false, false);
      bB[0] = Wc[(kl + kk    ) * WSTR + n0b + r16];
      bB[1] = Wc[(kl + kk + 1) * WSTR + n0b + r16];
      acc1 = __builtin_amdgcn_wmma_f32_16x16x4_f32(
          false, a, false, bB, (short)0, acc1, false, false);
    }

    wait_async0();     // next chunk landed (this wave's async ops)
    __syncthreads();   // ...and every other wave's too
  }

  const int mbase = (lane < 16) ? 0 : 8;
  const float bvA = bias[n0a + r16];
  const float bvB = bias[n0b + r16];
  #pragma unroll
  for (int i = 0; i < 8; ++i) {
    Y[(row0 + mbase + i) * DD + n0a + r16] = acc0[i] + bvA;
    Y[(row0 + mbase + i) * DD + n0b + r16] = acc1[i] + bvB;
  }
}

// ---------------------------------------------------------------------------
// Init segment state: m = -inf, z = 0, agg = 0.
// ---------------------------------------------------------------------------
__global__ __launch_bounds__(256)
void init_kernel(float* __restrict__ m, float* __restrict__ z,
                 float* __restrict__ agg, int nh, int nd) {
  const int i = blockIdx.x * 256 + threadIdx.x;
  if (i < nd) agg[i] = 0.0f;
  if (i < nh) {
    ((int*)m)[i] = 0xFF800000;  // -inf
    z[i] = 0.0f;
  }
}

// ---------------------------------------------------------------------------
// Float atomic max via int punning (correct for all signs given -inf init).
// ---------------------------------------------------------------------------
__device__ __forceinline__ void atomicMaxFloat(float* addr, float val) {
  if (val >= 0.0f) {
    atomicMax((int*)addr, __float_as_int(val));
  } else {
    atomicMin((unsigned int*)addr, (unsigned int)__float_as_int(val));
  }
}

// ---------------------------------------------------------------------------
// Per-edge attention scores + segment max. One wave per edge, 8 lanes/head.
// ---------------------------------------------------------------------------
__global__ __launch_bounds__(256)
void score_kernel(const float* __restrict__ q, const float* __restrict__ k,
                  const long long* __restrict__ ei,
                  float* __restrict__ scores, float* __restrict__ m) {
  const int e = blockIdx.x * 8 + (threadIdx.x >> 5);
  if (e >= EE) return;
  const int lane = threadIdx.x & 31;
  const long long src = ei[e];
  const long long dst = ei[EE + e];

  const float4* qp = (const float4*)(q + dst * DD + lane * 8);
  const float4* kp = (const float4*)(k + src * DD + lane * 8);
  const float4 q0 = qp[0], q1 = qp[1];
  const float4 k0 = kp[0], k1 = kp[1];

  float s = q0.x * k0.x + q0.y * k0.y + q0.z * k0.z + q0.w * k0.w
          + q1.x * k1.x + q1.y * k1.y + q1.z * k1.z + q1.w * k1.w;
  // reduce across the 8 lanes of one head
  s += __shfl_xor(s, 1, 8);
  s += __shfl_xor(s, 2, 8);
  s += __shfl_xor(s, 4, 8);
  s *= 0.125f;  // 1/sqrt(HD), HD=64

  if ((lane & 7) == 0) {
    const int h = lane >> 3;
    scores[e * HH + h] = s;
    atomicMaxFloat(&m[dst * HH + h], s);
  }
}

// ---------------------------------------------------------------------------
// e = exp(score - m[dst]); z[dst] += e. One thread per (edge, head).
// ---------------------------------------------------------------------------
__global__ __launch_bounds__(256)
void exp_kernel(const long long* __restrict__ ei, float* __restrict__ scores,
                const float* __restrict__ m, float* __restrict__ z) {
  const int t = blockIdx.x * 256 + threadIdx.x;
  if (t >= EE * HH) return;
  const int e = t >> 2;
  const int h = t & 3;
  const long long dst = ei[EE + e];
  const float ex = __expf(scores[t] - m[dst * HH + h]);
  scores[t] = ex;  // in-place (rewritten by score_kernel each call)
  atomicAdd(&z[dst * HH + h], ex);
}

// ---------------------------------------------------------------------------
// agg[dst] += (e/z[dst]) * v[src]. One wave per edge; 8 elems/lane, each
// lane's 8 elems stay within one head (8 | 64).
// ---------------------------------------------------------------------------
__global__ __launch_bounds__(256)
void agg_kernel(const float* __restrict__ v, const long long* __restrict__ ei,
                const float* __restrict__ esc, const float* __restrict__ z,
                float* __restrict__ agg) {
  const int e = blockIdx.x * 8 + (threadIdx.x >> 5);
  if (e >= EE) return;
  const int lane = threadIdx.x & 31;
  const long long src = ei[e];
  const long long dst = ei[EE + e];
  const int h = lane >> 3;

  const float alpha = esc[e * HH + h] / z[dst * HH + h];

  const float4* vp = (const float4*)(v + src * DD + lane * 8);
  const float4 v0 = vp[0], v1 = vp[1];
  float* ap = agg + dst * DD + lane * 8;
  atomicAdd(ap + 0, alpha * v0.x);
  atomicAdd(ap + 1, alpha * v0.y);
  atomicAdd(ap + 2, alpha * v0.z);
  atomicAdd(ap + 3, alpha * v0.w);
  atomicAdd(ap + 4, alpha * v1.x);
  atomicAdd(ap + 5, alpha * v1.y);
  atomicAdd(ap + 6, alpha * v1.z);
  atomicAdd(ap + 7, alpha * v1.w);
}

// ---------------------------------------------------------------------------
// Orchestration
// ---------------------------------------------------------------------------
extern "C" void kernel_launch(void* const* d_in, const int* in_sizes, int n_in,
                              void* d_out, int out_size, void* d_ws, size_t ws_size,
                              hipStream_t stream) {
  (void)in_sizes; (void)n_in; (void)out_size; (void)ws_size;

  const float*     x  = (const float*)d_in[0];
  const long long* ei = (const long long*)d_in[1];   // int64 [2, E]
  const float*     g1 = (const float*)d_in[2];
  const float*     b1 = (const float*)d_in[3];
  const float*     g2 = (const float*)d_in[4];
  const float*     b2 = (const float*)d_in[5];
  const float*     Wq = (const float*)d_in[6];
  const float*     bq = (const float*)d_in[7];
  const float*     Wk = (const float*)d_in[8];
  const float*     bk = (const float*)d_in[9];
  const float*     Wv = (const float*)d_in[10];
  const float*     bv = (const float*)d_in[11];
  const float*     Wo = (const float*)d_in[12];
  const float*     bo = (const float*)d_in[13];
  float* out = (float*)d_out;
  float* ws  = (float*)d_ws;

  // Workspace layout (floats). `oo` (pre-LN output) reuses xn's region
  // since xn is dead after the QKV GEMMs.
  float* xn = ws;                         // N*D  = 2,560,000
  float* q  = ws +  2560000;              // N*D
  float* k  = ws +  5120000;              // N*D
  float* v  = ws +  7680000;              // N*D
  float* sc = ws + 10240000;              // E*H  = 1,280,000
  float* mx = ws + 11520000;              // N*H  = 40,000
  float* zz = ws + 11560000;              // N*H
  float* ag = ws + 11600000;              // N*D
  float* oo = ws;                         // alias xn (dead)

  ln1_kernel<<<NN, 256, 0, stream>>>(x, g1, b1, xn);

  gemm16_kernel<<<NN / 16, 256, 0, stream>>>(xn, Wq, bq, q);
  gemm16_kernel<<<NN / 16, 256, 0, stream>>>(xn, Wk, bk, k);
  gemm16_kernel<<<NN / 16, 256, 0, stream>>>(xn, Wv, bv, v);

  init_kernel<<<(NN * DD + 255) / 256, 256, 0, stream>>>(mx, zz, ag,
                                                         NN * HH, NN * DD);

  score_kernel<<<EE / 8, 256, 0, stream>>>(q, k, ei, sc, mx);
  exp_kernel<<<(EE * HH + 255) / 256, 256, 0, stream>>>(ei, sc, mx, zz);
  agg_kernel<<<EE / 8, 256, 0, stream>>>(v, ei, sc, zz, ag);

  gemm16_kernel<<<NN / 16, 256, 0, stream>>>(ag, Wo, bo, oo);

  ln2_kernel<<<NN, 256, 0, stream>>>(oo, x, g2, b2, out);
}